// GCN_22857815949368
// MI455X (gfx1250) — compile-verified
//
#include <hip/hip_runtime.h>

typedef __attribute__((ext_vector_type(2))) float v2f;
typedef __attribute__((ext_vector_type(8))) float v8f;

#define NN   100000
#define EE   1600000
#define FIN  128
#define FH   64

// ---------------------------------------------------------------------------
// Degree / normalization
// ---------------------------------------------------------------------------
__global__ void init_deg_k(float* __restrict__ deg) {
    int i = blockIdx.x * blockDim.x + threadIdx.x;
    if (i < NN) deg[i] = 1.0f;                      // self-loop contributes 1
}

__global__ void count_deg_k(float* __restrict__ deg, const int* __restrict__ col) {
    int e = blockIdx.x * blockDim.x + threadIdx.x;
    if (e < EE) atomicAdd(&deg[col[e]], 1.0f);
}

__global__ void dinv_k(float* __restrict__ deg) {
    int i = blockIdx.x * blockDim.x + threadIdx.x;
    if (i < NN) deg[i] = rsqrtf(deg[i]);            // deg >= 1 always
}

// ---------------------------------------------------------------------------
// GEMM: Out[nrows x 64] = X[nrows x K] @ W[K x 64], fp32 via V_WMMA_F32_16X16X4_F32
// One wave computes a 16-row x 64-col tile. Wave-uniform exit keeps EXEC all-1s.
// ---------------------------------------------------------------------------
template <int K>
__global__ __launch_bounds__(256) void gemm_wmma_k(const float* __restrict__ X,
                                                   const float* __restrict__ W,
                                                   float* __restrict__ Out,
                                                   int nrows) {
    const int wave = blockIdx.x * (256 / 32) + (threadIdx.x >> 5);
    const int lane = threadIdx.x & 31;
    const int row0 = wave << 4;
    if (row0 >= nrows) return;                       // wave-uniform

    const int m  = lane & 15;                        // A row / B col within tile
    const int kh = (lane >> 4) << 1;                 // K sub-offset: 0 or 2

    const float* xr = X + (size_t)(row0 + m) * K;

    v8f acc0 = {}, acc1 = {}, acc2 = {}, acc3 = {};

    for (int k0 = 0; k0 < K; k0 += 4) {
        // A fragment: lane holds A[m][k0+kh], A[m][k0+kh+1]  (8B-aligned)
        v2f a = *(const v2f*)(xr + k0 + kh);

        const float* w0 = W + (size_t)(k0 + kh) * FH + m;   // B[kh][m + n0]
        const float* w1 = w0 + FH;                           // B[kh+1][m + n0]
        v2f b;
        b.x = w0[0];  b.y = w1[0];
        acc0 = __builtin_amdgcn_wmma_f32_16x16x4_f32(false, a, false, b,
                                                     (short)0, acc0, false, false);
        b.x = w0[16]; b.y = w1[16];
        acc1 = __builtin_amdgcn_wmma_f32_16x16x4_f32(false, a, false, b,
                                                     (short)0, acc1, false, false);
        b.x = w0[32]; b.y = w1[32];
        acc2 = __builtin_amdgcn_wmma_f32_16x16x4_f32(false, a, false, b,
                                                     (short)0, acc2, false, false);
        b.x = w0[48]; b.y = w1[48];
        acc3 = __builtin_amdgcn_wmma_f32_16x16x4_f32(false, a, false, b,
                                                     (short)0, acc3, false, false);
    }

    // C/D layout: VGPR v, lanes 0-15 -> row v, lanes 16-31 -> row v+8; col = lane&15
    const int rbase = row0 + ((lane >> 4) << 3);
    float* o = Out + (size_t)rbase * FH + m;
#pragma unroll
    for (int v = 0; v < 8; ++v) {
        o[(size_t)v * FH +  0] = acc0[v];
        o[(size_t)v * FH + 16] = acc1[v];
        o[(size_t)v * FH + 32] = acc2[v];
        o[(size_t)v * FH + 48] = acc3[v];
    }
}

// ---------------------------------------------------------------------------
// hs = h * dinv[node]; acc initialized to hs (covers the self-loop edge)
// ---------------------------------------------------------------------------
__global__ void scale_init_k(const float* __restrict__ H, const float* __restrict__ dinv,
                             float* __restrict__ HS, float* __restrict__ ACC) {
    int i = blockIdx.x * blockDim.x + threadIdx.x;
    if (i >= NN * FH) return;
    float v = H[i] * dinv[i >> 6];
    HS[i]  = v;
    ACC[i] = v;
}

// ---------------------------------------------------------------------------
// acc[col] += hs[row] over the E original edges. 16 lanes per edge, float4 each:
// 256B row gathers fully coalesced; per-float global_atomic_add_f32 scatter (L2-resident).
// ---------------------------------------------------------------------------
__global__ void scatter_k(const float* __restrict__ HS, float* __restrict__ ACC,
                          const int* __restrict__ row, const int* __restrict__ col) {
    int t = blockIdx.x * blockDim.x + threadIdx.x;
    if (t >= EE * 16) return;
    int e = t >> 4;
    int c = (t & 15) << 2;
    int r  = row[e];
    int cl = col[e];
    float4 v = *(const float4*)(HS + (size_t)r * FH + c);
    float* dst = ACC + (size_t)cl * FH + c;
    atomicAdd(dst + 0, v.x);
    atomicAdd(dst + 1, v.y);
    atomicAdd(dst + 2, v.z);
    atomicAdd(dst + 3, v.w);
}

// ---------------------------------------------------------------------------
// out = relu(acc * dinv[node] + bias[f])
// ---------------------------------------------------------------------------
__global__ void post_k(const float* __restrict__ ACC, const float* __restrict__ dinv,
                       const float* __restrict__ bias, float* __restrict__ Out) {
    int i = blockIdx.x * blockDim.x + threadIdx.x;
    if (i >= NN * FH) return;
    float v = ACC[i] * dinv[i >> 6] + bias[i & 63];
    Out[i] = v > 0.0f ? v : 0.0f;
}

// ---------------------------------------------------------------------------
// out[e] = dot(h[row]*h[col], Wout) + bout — 16 lanes/edge, shfl reduction
// ---------------------------------------------------------------------------
__global__ void edge_out_k(const float* __restrict__ H, const int* __restrict__ row,
                           const int* __restrict__ col, const float* __restrict__ Wout,
                           const float* __restrict__ bout, float* __restrict__ out) {
    int t = blockIdx.x * blockDim.x + threadIdx.x;
    if (t >= EE * 16) return;
    int e = t >> 4;
    int c = (t & 15) << 2;
    int r  = row[e];
    int cl = col[e];
    float4 a = *(const float4*)(H + (size_t)r  * FH + c);
    float4 b = *(const float4*)(H + (size_t)cl * FH + c);
    float4 w = *(const float4*)(Wout + c);
    float p = a.x * b.x * w.x + a.y * b.y * w.y + a.z * b.z * w.z + a.w * b.w * w.w;
#pragma unroll
    for (int off = 8; off >= 1; off >>= 1)
        p += __shfl_down(p, off, 16);
    if ((t & 15) == 0) out[e] = p + bout[0];
}

// ---------------------------------------------------------------------------
extern "C" void kernel_launch(void* const* d_in, const int* in_sizes, int n_in,
                              void* d_out, int out_size, void* d_ws, size_t ws_size,
                              hipStream_t stream) {
    (void)in_sizes; (void)n_in; (void)out_size; (void)ws_size;

    const float* x    = (const float*)d_in[0];
    const int*   ei   = (const int*)d_in[1];     // [2, E] flat
    const int*   row  = ei;
    const int*   col  = ei + EE;
    const float* W1   = (const float*)d_in[2];
    const float* b1   = (const float*)d_in[3];
    const float* W2   = (const float*)d_in[4];
    const float* b2   = (const float*)d_in[5];
    const float* Wout = (const float*)d_in[6];
    const float* bout = (const float*)d_in[7];
    float* out = (float*)d_out;

    // Workspace layout (~77.2 MB)
    float* dinv = (float*)d_ws;                        // N (reused: deg -> dinv)
    size_t npad = (size_t)((NN + 255) & ~255);
    float* bufA = dinv + npad;                         // N*64
    float* bufB = bufA + (size_t)NN * FH;              // N*64
    float* bufC = bufB + (size_t)NN * FH;              // N*64

    const int TB = 256;
    const int gN   = (NN + TB - 1) / TB;
    const int gE   = (EE + TB - 1) / TB;
    const int gNF  = (NN * FH + TB - 1) / TB;
    const int gE16 = (EE * 16) / TB;                   // 25.6M / 256 exact
    const int gGemm = (NN / 16 + 7) / 8;               // 6250 waves, 8 waves/block

    // Degrees (with self-loops) -> dinv
    init_deg_k <<<gN, TB, 0, stream>>>(dinv);
    count_deg_k<<<gE, TB, 0, stream>>>(dinv, col);
    dinv_k     <<<gN, TB, 0, stream>>>(dinv);

    // Layer 1: h1 = relu( D^-1/2 A D^-1/2 (x @ W1) + b1 )
    gemm_wmma_k<FIN><<<gGemm, TB, 0, stream>>>(x, W1, bufA, NN);
    scale_init_k<<<gNF, TB, 0, stream>>>(bufA, dinv, bufB, bufC);
    scatter_k   <<<gE16, TB, 0, stream>>>(bufB, bufC, row, col);
    post_k      <<<gNF, TB, 0, stream>>>(bufC, dinv, b1, bufA);

    // Layer 2: h2 = relu( D^-1/2 A D^-1/2 (h1 @ W2) + b2 )
    gemm_wmma_k<FH><<<gGemm, TB, 0, stream>>>(bufA, W2, bufB, NN);
    scale_init_k<<<gNF, TB, 0, stream>>>(bufB, dinv, bufC, bufA);
    scatter_k   <<<gE16, TB, 0, stream>>>(bufC, bufA, row, col);
    post_k      <<<gNF, TB, 0, stream>>>(bufA, dinv, b2, bufB);

    // Edge scoring on original edges
    edge_out_k<<<gE16, TB, 0, stream>>>(bufB, row, col, Wout, bout, out);
}